// UniformFSQ_32315333935396
// MI455X (gfx1250) — compile-verified
//
#include <hip/hip_runtime.h>

// CDNA5 / gfx1250. wave32. Uniform FSQ: quantize to 8 levels per channel,
// emit quantized values + indices + commitment loss.
// Streaming kernel is bandwidth-bound (302 MB @ 23.3 TB/s ~ 13 us floor):
// float4 in -> float4 q + int4 idx out, f32 loss accumulator.
// Block-level loss reduction uses V_WMMA_F32_16X16X4_F32 (ones x partials)
// to collapse 64 partials -> 16 column sums in one matrix op, then shfl_xor.

typedef __attribute__((ext_vector_type(2))) float v2f;
typedef __attribute__((ext_vector_type(8))) float v8f;

#define FSQ_HALF 3.5f
#define FSQ_INVH (1.0f / 3.5f)

__device__ __forceinline__ void quant1(float xin, float& qo, int& io, float& acc) {
    float xc = fminf(fmaxf(xin, -1.0f), 1.0f);
    float t  = (xc + 1.0f) * FSQ_HALF;      // in [0, 7]
    float f  = rintf(t);                     // round-half-to-even == jnp.round
    float q  = f * FSQ_INVH - 1.0f;
    float d  = xin - q;                      // loss uses UNclipped x
    acc = fmaf(d, d, acc);
    qo = q;
    io = (int)f;                             // exact, f in {0..7}
}

__global__ __launch_bounds__(256) void fsq_main_kernel(
    const float* __restrict__ x,
    float* __restrict__ outq,
    int*   __restrict__ outi,
    float* __restrict__ partial,
    long long n4, long long N)
{
    const long long tid    = (long long)blockIdx.x * blockDim.x + threadIdx.x;
    const long long stride = (long long)gridDim.x * blockDim.x;

    const float4* __restrict__ xv = (const float4*)x;
    float4* __restrict__ qv = (float4*)outq;
    int4*   __restrict__ iv = (int4*)outi;

    float lsum = 0.0f;

    for (long long i = tid; i < n4; i += stride) {
        float4 v = xv[i];                    // global_load_b128
        float4 q; int4 k;
        quant1(v.x, q.x, k.x, lsum);
        quant1(v.y, q.y, k.y, lsum);
        quant1(v.z, q.z, k.z, lsum);
        quant1(v.w, q.w, k.w, lsum);
        qv[i] = q;                           // global_store_b128
        iv[i] = k;                           // global_store_b128
    }

    // Scalar tail (N % 4 != 0), handled by block 0 only.
    long long base = n4 * 4;
    if (blockIdx.x == 0 && (long long)threadIdx.x < (N - base)) {
        long long j = base + threadIdx.x;
        float q; int k;
        quant1(x[j], q, k, lsum);
        outq[j] = q;
        outi[j] = k;
    }

    // ---- Block reduction of lsum via V_WMMA_F32_16X16X4_F32 ----
    __shared__ float smem[256];
    smem[threadIdx.x] = lsum;
    __syncthreads();

    // Every wave builds the SAME B operand from all 256 partials, so EXEC is
    // all-1s for the WMMA in every wave (no divergence around the matrix op).
    const int lane = threadIdx.x & 31;
    float p0 = smem[lane * 8 + 0] + smem[lane * 8 + 1] +
               smem[lane * 8 + 2] + smem[lane * 8 + 3];
    float p1 = smem[lane * 8 + 4] + smem[lane * 8 + 5] +
               smem[lane * 8 + 6] + smem[lane * 8 + 7];

    // A = ones(16x4), B = partials(4x16): D[m][n] = sum_k B[k][n].
    v2f a; a[0] = 1.0f; a[1] = 1.0f;
    v2f b; b[0] = p0;   b[1] = p1;
    v8f c = {};
    c = __builtin_amdgcn_wmma_f32_16x16x4_f32(
            /*neg_a=*/false, a, /*neg_b=*/false, b,
            /*c_mod=*/(short)0, c, /*reuse_a=*/false, /*reuse_b=*/false);

    // c[0]: lanes 0..15 hold D[0][n] (column sums); lanes 16..31 duplicate
    // them as D[8][n]. Reduce the 16 column sums within each half-wave.
    float s = c[0];
    s += __shfl_xor(s, 8, 32);
    s += __shfl_xor(s, 4, 32);
    s += __shfl_xor(s, 2, 32);
    s += __shfl_xor(s, 1, 32);

    if (threadIdx.x == 0) partial[blockIdx.x] = s;   // lane 0 of wave 0
}

__global__ __launch_bounds__(256) void fsq_reduce_kernel(
    const float* __restrict__ partial, int nblocks,
    float* __restrict__ lossOut, float scale)
{
    __shared__ float smem[256];
    float s = 0.0f;
    for (int i = threadIdx.x; i < nblocks; i += 256) s += partial[i];
    smem[threadIdx.x] = s;
    __syncthreads();
    for (int off = 128; off > 0; off >>= 1) {
        if ((int)threadIdx.x < off) smem[threadIdx.x] += smem[threadIdx.x + off];
        __syncthreads();
    }
    if (threadIdx.x == 0) *lossOut = smem[0] * scale;   // deterministic order
}

extern "C" void kernel_launch(void* const* d_in, const int* in_sizes, int n_in,
                              void* d_out, int out_size, void* d_ws, size_t ws_size,
                              hipStream_t stream) {
    const float* x = (const float*)d_in[0];
    const long long N  = (long long)in_sizes[0];   // 25,165,824
    const long long n4 = N / 4;

    float* outq = (float*)d_out;                   // [0, N)       quantized_st
    int*   outi = (int*)(outq + N);                // [N, 2N)      indices (i32 bits)
    float* loss = (float*)(outq + 2 * N);          // [2N]         scalar loss

    float* partial = (float*)d_ws;

    int blocks = 3072;                              // 8 float4 iters/thread at this N
    long long cap = (long long)(ws_size / sizeof(float));
    if (cap < blocks) blocks = (cap > 0) ? (int)cap : 1;

    fsq_main_kernel<<<blocks, 256, 0, stream>>>(x, outq, outi, partial, n4, N);
    fsq_reduce_kernel<<<1, 256, 0, stream>>>(partial, blocks, loss,
                                             0.25f / (float)N);
}